// HGINLayer_38543036514754
// MI455X (gfx1250) — compile-verified
//
#include <hip/hip_runtime.h>
#include <hip/hip_bf16.h>

typedef __attribute__((ext_vector_type(2))) float v2f;
typedef __attribute__((ext_vector_type(8))) float v8f;

#define D 128
#define BN_EPS 1e-5f

// ---------------------------------------------------------------------------
// zero fill
// ---------------------------------------------------------------------------
__global__ void zero_f32(float* __restrict__ p, long long n) {
  long long i = (long long)blockIdx.x * blockDim.x + threadIdx.x;
  long long stride = (long long)gridDim.x * blockDim.x;
  for (; i < n; i += stride) p[i] = 0.0f;
}

// ---------------------------------------------------------------------------
// scatter-add: accum[dst[e], :] += feat[src[e], :]   (one wave32 per edge,
// each lane moves 4 consecutive floats -> float4 gather + 4 f32 atomics).
// accum (51 MB) is L2-resident on the 192 MB L2, so these atomics stay on-chip.
// ---------------------------------------------------------------------------
__global__ void scatter_add_rows(const float* __restrict__ feat,
                                 const int* __restrict__ src,
                                 const int* __restrict__ dst,
                                 float* __restrict__ accum, int nedges) {
  const int wave = blockIdx.x * (blockDim.x >> 5) + (threadIdx.x >> 5);
  const int lane = threadIdx.x & 31;
  if (wave >= nedges) return;
  const int s = src[wave];
  const int d = dst[wave];
  const float4 v = *(const float4*)(feat + (size_t)s * D + lane * 4);
  float* o = accum + (size_t)d * D + lane * 4;
  atomicAdd(o + 0, v.x);
  atomicAdd(o + 1, v.y);
  atomicAdd(o + 2, v.z);
  atomicAdd(o + 3, v.w);
}

// ---------------------------------------------------------------------------
// H = (alpha * X + A) @ W + bias          (alpha = 1 + eps, from device scalar)
// optionally accumulates per-column sum / sum-of-squares into stats[0..2D)
// One 256-thread block = 8 waves; wave w computes rows [blk*128 + 16w, +16),
// all 128 output columns, via V_WMMA_F32_16X16X4_F32 (K-loop of 32 steps).
// Epilogue has a wave-uniform full-tile fast path (no per-store exec masking).
// ---------------------------------------------------------------------------
template <bool HAS_ADD, bool DO_STATS>
__global__ __launch_bounds__(256)
void gemm_tile(const float* __restrict__ X, const float* __restrict__ A,
               const float* __restrict__ eps_p,
               const float* __restrict__ W, const float* __restrict__ bias,
               float* __restrict__ H, float* __restrict__ stats, int nrows) {
  __shared__ float sW[D * D];
  __shared__ float s_sum[D];
  __shared__ float s_sq[D];

  for (int i = threadIdx.x; i < D * D / 4; i += blockDim.x)
    ((float4*)sW)[i] = ((const float4*)W)[i];
  if (DO_STATS && threadIdx.x < D) {
    s_sum[threadIdx.x] = 0.0f;
    s_sq[threadIdx.x] = 0.0f;
  }
  __syncthreads();

  const int lane = threadIdx.x & 31;
  const int wv   = threadIdx.x >> 5;
  const int half = lane >> 4;   // 0: K=k0..k0+1, 1: K=k0+2..k0+3
  const int l16  = lane & 15;
  const int row0 = blockIdx.x * 128 + wv * 16;

  int arow = row0 + l16;                    // A-fragment row for this lane
  if (arow > nrows - 1) arow = nrows - 1;   // clamp (WMMA needs full EXEC)

  const float alpha = HAS_ADD ? (1.0f + eps_p[0]) : 1.0f;

  v8f acc[8] = {};  // 8 N-tiles of 16 columns

  for (int k0 = 0; k0 < D; k0 += 4) {
    const int kk = k0 + 2 * half;
    // A fragment: 16x4 f32 tile of Y = alpha*X + A
    v2f x2 = *(const v2f*)(X + (size_t)arow * D + kk);
    if (HAS_ADD) {
      const v2f a2 = *(const v2f*)(A + (size_t)arow * D + kk);
      x2 = alpha * x2 + a2;
    }
#pragma unroll
    for (int n = 0; n < 8; ++n) {
      // B fragment: 4x16 f32 tile of W (row-major K x N) from LDS
      v2f b2;
      b2.x = sW[kk * D + n * 16 + l16];
      b2.y = sW[(kk + 1) * D + n * 16 + l16];
      acc[n] = __builtin_amdgcn_wmma_f32_16x16x4_f32(
          false, x2, false, b2, (short)0, acc[n], false, false);
    }
  }

  // epilogue: bias, store, per-block BN stats
  if (row0 + 16 <= nrows) {
    // full tile: unconditional coalesced stores, unmasked stats
#pragma unroll
    for (int n = 0; n < 8; ++n) {
      const int col = n * 16 + l16;
      const float bcol = bias[col];
      float psum = 0.0f, psq = 0.0f;
#pragma unroll
      for (int r = 0; r < 8; ++r) {
        const int row = row0 + r + half * 8;  // C layout: VGPR r -> M=r / M=r+8
        const float v = acc[n][r] + bcol;
        H[(size_t)row * D + col] = v;
        if (DO_STATS) { psum += v; psq += v * v; }
      }
      if (DO_STATS) {
        atomicAdd(&s_sum[col], psum);   // ds_add_f32 (block-local reduce)
        atomicAdd(&s_sq[col], psq);
      }
    }
  } else {
    // tail tile (only the last block): masked stores/stats
#pragma unroll
    for (int n = 0; n < 8; ++n) {
      const int col = n * 16 + l16;
      const float bcol = bias[col];
      float psum = 0.0f, psq = 0.0f;
#pragma unroll
      for (int r = 0; r < 8; ++r) {
        const int row = row0 + r + half * 8;
        const float v = acc[n][r] + bcol;
        if (row < nrows) {
          H[(size_t)row * D + col] = v;
          if (DO_STATS) { psum += v; psq += v * v; }
        }
      }
      if (DO_STATS) {
        atomicAdd(&s_sum[col], psum);
        atomicAdd(&s_sq[col], psq);
      }
    }
  }
  if (DO_STATS) {
    __syncthreads();
    if (threadIdx.x < D) {
      atomicAdd(stats + threadIdx.x, s_sum[threadIdx.x]);
      atomicAdd(stats + D + threadIdx.x, s_sq[threadIdx.x]);
    }
  }
}

// ---------------------------------------------------------------------------
// stats -> (scale, shift): scale = g * rsqrt(var + eps); shift = beta - mu*scale
// stored at stats[2D..3D) and stats[3D..4D)
// ---------------------------------------------------------------------------
__global__ void bn_finalize(float* __restrict__ stats, const float* __restrict__ g,
                            const float* __restrict__ beta, int nrows) {
  const int c = threadIdx.x;
  if (c < D) {
    const float inv = 1.0f / (float)nrows;
    const float mu = stats[c] * inv;
    const float var = stats[D + c] * inv - mu * mu;   // biased var (matches ref)
    const float rs = rsqrtf(var + BN_EPS);
    const float sc = g[c] * rs;
    stats[2 * D + c] = sc;
    stats[3 * D + c] = beta[c] - mu * sc;
  }
}

// ---------------------------------------------------------------------------
// Out = relu(scale*H + shift) @ W + bias     (BN+ReLU fused into A-frag load)
// ---------------------------------------------------------------------------
__global__ __launch_bounds__(256)
void bn_relu_gemm(const float* __restrict__ Hin, const float* __restrict__ scale,
                  const float* __restrict__ shift,
                  const float* __restrict__ W, const float* __restrict__ bias,
                  float* __restrict__ Out, int nrows) {
  __shared__ float sW[D * D];
  __shared__ float s_sc[D];
  __shared__ float s_sh[D];

  for (int i = threadIdx.x; i < D * D / 4; i += blockDim.x)
    ((float4*)sW)[i] = ((const float4*)W)[i];
  if (threadIdx.x < D) {
    s_sc[threadIdx.x] = scale[threadIdx.x];
    s_sh[threadIdx.x] = shift[threadIdx.x];
  }
  __syncthreads();

  const int lane = threadIdx.x & 31;
  const int wv   = threadIdx.x >> 5;
  const int half = lane >> 4;
  const int l16  = lane & 15;
  const int row0 = blockIdx.x * 128 + wv * 16;

  int arow = row0 + l16;
  if (arow > nrows - 1) arow = nrows - 1;

  v8f acc[8] = {};

  for (int k0 = 0; k0 < D; k0 += 4) {
    const int kk = k0 + 2 * half;
    const v2f h2 = *(const v2f*)(Hin + (size_t)arow * D + kk);
    v2f x2;
    x2.x = fmaxf(h2.x * s_sc[kk] + s_sh[kk], 0.0f);
    x2.y = fmaxf(h2.y * s_sc[kk + 1] + s_sh[kk + 1], 0.0f);
#pragma unroll
    for (int n = 0; n < 8; ++n) {
      v2f b2;
      b2.x = sW[kk * D + n * 16 + l16];
      b2.y = sW[(kk + 1) * D + n * 16 + l16];
      acc[n] = __builtin_amdgcn_wmma_f32_16x16x4_f32(
          false, x2, false, b2, (short)0, acc[n], false, false);
    }
  }

  if (row0 + 16 <= nrows) {
#pragma unroll
    for (int n = 0; n < 8; ++n) {
      const int col = n * 16 + l16;
      const float bcol = bias[col];
#pragma unroll
      for (int r = 0; r < 8; ++r) {
        const int row = row0 + r + half * 8;
        Out[(size_t)row * D + col] = acc[n][r] + bcol;
      }
    }
  } else {
#pragma unroll
    for (int n = 0; n < 8; ++n) {
      const int col = n * 16 + l16;
      const float bcol = bias[col];
#pragma unroll
      for (int r = 0; r < 8; ++r) {
        const int row = row0 + r + half * 8;
        if (row < nrows) Out[(size_t)row * D + col] = acc[n][r] + bcol;
      }
    }
  }
}

static inline size_t imax_sz(size_t a, size_t b) { return a > b ? a : b; }

extern "C" void kernel_launch(void* const* d_in, const int* in_sizes, int n_in,
                              void* d_out, int out_size, void* d_ws, size_t ws_size,
                              hipStream_t stream) {
  const float* feat_target = (const float*)d_in[0];
  const float* feat_other  = (const float*)d_in[1];
  const int*   go_src  = (const int*)d_in[2];
  const int*   go_dst  = (const int*)d_in[3];
  const int*   ret_src = (const int*)d_in[4];
  const int*   ret_dst = (const int*)d_in[5];
  const float* go_W1  = (const float*)d_in[6];
  const float* go_b1  = (const float*)d_in[7];
  const float* go_g   = (const float*)d_in[8];
  const float* go_be  = (const float*)d_in[9];
  const float* go_W2  = (const float*)d_in[10];
  const float* go_b2  = (const float*)d_in[11];
  const float* ret_W1 = (const float*)d_in[12];
  const float* ret_b1 = (const float*)d_in[13];
  const float* ret_g  = (const float*)d_in[14];
  const float* ret_be = (const float*)d_in[15];
  const float* ret_W2 = (const float*)d_in[16];
  const float* ret_b2 = (const float*)d_in[17];
  const float* proj_W = (const float*)d_in[18];
  const float* proj_b = (const float*)d_in[19];
  const float* eps_go  = (const float*)d_in[20];
  const float* eps_ret = (const float*)d_in[21];

  const int NT = in_sizes[0] / D;   // 100000 target nodes
  const int NO = in_sizes[1] / D;   // 100000 other nodes
  const int E  = in_sizes[2];       // 1.6M edges per etype

  float* h_target = (float*)d_out;                        // NT x D
  float* h_other  = (float*)d_out + (size_t)NT * D;       // NO x D

  const size_t nmax = imax_sz((size_t)NT, (size_t)NO) * D;
  float* accum  = (float*)d_ws;            // reused: a_other then a_target
  float* Hbuf   = accum + nmax;            // MLP hidden activations
  float* Pbuf   = Hbuf + nmax;             // projected target features
  float* statsA = Pbuf + (size_t)NT * D;   // [sum, sumsq, scale, shift] go
  float* statsB = statsA + 4 * D;          // same for ret

  const int nob = (NO + 127) / 128;
  const int ntb = (NT + 127) / 128;
  const int sab = (E + 7) / 8;             // 8 edge-waves per 256-thread block

  // ---- go path:  a_other = segsum(feat_target[go_src] -> go_dst) ----------
  zero_f32<<<1024, 256, 0, stream>>>(accum, (long long)NO * D);
  zero_f32<<<1, 256, 0, stream>>>(statsA, 8 * D);   // both stats blocks
  scatter_add_rows<<<sab, 256, 0, stream>>>(feat_target, go_src, go_dst, accum, E);

  // h_other = MLP_go((1+eps_go)*feat_other + a_other)
  gemm_tile<true, true><<<nob, 256, 0, stream>>>(
      feat_other, accum, eps_go, go_W1, go_b1, Hbuf, statsA, NO);
  bn_finalize<<<1, 128, 0, stream>>>(statsA, go_g, go_be, NO);
  bn_relu_gemm<<<nob, 256, 0, stream>>>(
      Hbuf, statsA + 2 * D, statsA + 3 * D, go_W2, go_b2, h_other, NO);

  // ---- return path: a_target = segsum(h_other[ret_src] -> ret_dst) --------
  zero_f32<<<1024, 256, 0, stream>>>(accum, (long long)NT * D);
  scatter_add_rows<<<sab, 256, 0, stream>>>(h_other, ret_src, ret_dst, accum, E);

  // P = feat_target @ proj_W + proj_b
  gemm_tile<false, false><<<ntb, 256, 0, stream>>>(
      feat_target, nullptr, nullptr, proj_W, proj_b, Pbuf, nullptr, NT);

  // h_target = MLP_ret((1+eps_ret)*P + a_target)
  gemm_tile<true, true><<<ntb, 256, 0, stream>>>(
      Pbuf, accum, eps_ret, ret_W1, ret_b1, Hbuf, statsB, NT);
  bn_finalize<<<1, 128, 0, stream>>>(statsB, ret_g, ret_be, NT);
  bn_relu_gemm<<<ntb, 256, 0, stream>>>(
      Hbuf, statsB + 2 * D, statsB + 3 * D, ret_W2, ret_b2, h_target, NT);
}